// TopKRouter_53231824666802
// MI455X (gfx1250) — compile-verified
//
#include <hip/hip_runtime.h>
#include <hip/hip_bf16.h>

typedef __attribute__((ext_vector_type(2))) float v2f;
typedef __attribute__((ext_vector_type(4))) float v4f;
typedef __attribute__((ext_vector_type(8))) float v8f;

#define HID     4096
#define NEXP    64
#define TOPK    8
#define TILE_T  16
#define LDS_PITCH 65   // 64 + 1 padding to avoid 64-bank conflicts

// ---------------------------------------------------------------------------
// Stage 1: one workgroup per 16-token tile.
//   4 waves x (16 tokens x 16 experts) fp32 WMMA -> logits in LDS ->
//   softmax + top-8 + renorm -> outputs; deterministic per-WG aux partials.
//
// Inner loop uses b128 loads: per 8 K-values, each lane loads 4 consecutive
// floats of A and of B at column offset 4*half, and two WMMAs consume the
// {0,1} and {2,3} element pairs. Since WMMA's K-slot -> lane mapping is the
// same for A and B, remapping global K across slots is accumulation-safe.
// ---------------------------------------------------------------------------
__global__ __launch_bounds__(128) void router_stage1(
    const float* __restrict__ H,      // (T, 4096)
    const float* __restrict__ G,      // (64, 4096)
    float* __restrict__ out_w,        // (T, 8) weights
    float* __restrict__ out_i,        // (T, 8) indices (as float values)
    float* __restrict__ ws_psum,      // (nWG, 64) partial prob sums
    float* __restrict__ ws_cnt)       // (nWG, 64) partial top-k counts
{
    __shared__ float probs[TILE_T * LDS_PITCH];
    __shared__ int   cnt[NEXP];

    const int tid   = threadIdx.x;
    const int wave  = tid >> 5;       // 0..3 -> expert block
    const int lane  = tid & 31;
    const int half  = lane >> 4;      // K-slot group: lanes 16-31 = slots {2,3}
    const int col   = lane & 15;      // M row (A) / N col (B)
    const int tokenBase  = blockIdx.x * TILE_T;
    const int expertBase = wave * 16;

    if (tid < NEXP) cnt[tid] = 0;

    // Each lane streams 16B chunks: half 0 at +0, half 1 at +16 bytes.
    const float* __restrict__ Arow = H + (size_t)(tokenBase  + col) * HID + 4 * half;
    const float* __restrict__ Brow = G + (size_t)(expertBase + col) * HID + 4 * half;

    v8f c = {0.f, 0.f, 0.f, 0.f, 0.f, 0.f, 0.f, 0.f};
#pragma unroll 4
    for (int k = 0; k < HID; k += 8) {
        v4f a = *(const v4f*)(Arow + k);     // global_load_b128
        v4f b = *(const v4f*)(Brow + k);     // global_load_b128
        v2f a0 = __builtin_shufflevector(a, a, 0, 1);
        v2f a1 = __builtin_shufflevector(a, a, 2, 3);
        v2f b0 = __builtin_shufflevector(b, b, 0, 1);
        v2f b1 = __builtin_shufflevector(b, b, 2, 3);
        // WMMA #1: global K in {k,k+1} (lanes<16) U {k+4,k+5} (lanes>=16)
        c = __builtin_amdgcn_wmma_f32_16x16x4_f32(
                false, a0, false, b0, (short)0, c, false, false);
        // WMMA #2: global K in {k+2,k+3} U {k+6,k+7}
        c = __builtin_amdgcn_wmma_f32_16x16x4_f32(
                false, a1, false, b1, (short)0, c, false, false);
    }

    // C/D layout: VGPR r, lanes 0-15 -> M=r, N=lane; lanes 16-31 -> M=r+8, N=lane-16
#pragma unroll
    for (int r = 0; r < 8; ++r) {
        probs[(half * 8 + r) * LDS_PITCH + expertBase + col] = c[r];
    }
    __syncthreads();

    // ---- per-token softmax + top-8 (thread t owns token t) ----
    if (tid < TILE_T) {
        float* __restrict__ rowp = &probs[tid * LDS_PITCH];

        float m = rowp[0];
#pragma unroll
        for (int e = 1; e < NEXP; ++e) m = fmaxf(m, rowp[e]);

        float s = 0.f;
#pragma unroll
        for (int e = 0; e < NEXP; ++e) {
            float p = __expf(rowp[e] - m);
            rowp[e] = p;
            s += p;
        }
        const float inv = 1.0f / s;
#pragma unroll
        for (int e = 0; e < NEXP; ++e) rowp[e] *= inv;   // softmax probs

        // top-8, descending, lowest index wins ties (strict > scan)
        float vals[TOPK];
        int   idxs[TOPK];
        unsigned long long taken = 0ull;
        float wsum = 0.f;
#pragma unroll
        for (int j = 0; j < TOPK; ++j) {
            float best = -1.0f;
            int   bi   = 0;
            for (int e = 0; e < NEXP; ++e) {
                bool free = ((taken >> e) & 1ull) == 0ull;
                float v = rowp[e];
                if (free && v > best) { best = v; bi = e; }
            }
            taken |= (1ull << bi);
            vals[j] = best;
            idxs[j] = bi;
            wsum += best;
            atomicAdd(&cnt[bi], 1);   // LDS integer atomic: order-insensitive
        }

        const float invw = 1.0f / wsum;
        const size_t t = (size_t)tokenBase + tid;
#pragma unroll
        for (int j = 0; j < TOPK; ++j) {
            out_w[t * TOPK + j] = vals[j] * invw;
            out_i[t * TOPK + j] = (float)idxs[j];
        }
    }
    __syncthreads();

    // ---- deterministic aux-loss partials: thread e sums its prob column ----
    if (tid < NEXP) {
        float ps = 0.f;
#pragma unroll
        for (int t = 0; t < TILE_T; ++t) ps += probs[t * LDS_PITCH + tid];
        ws_psum[(size_t)blockIdx.x * NEXP + tid] = ps;
        ws_cnt [(size_t)blockIdx.x * NEXP + tid] = (float)cnt[tid];
    }
}

// ---------------------------------------------------------------------------
// Stage 2: fixed-order reduction of the per-WG partials -> aux loss scalar.
// ---------------------------------------------------------------------------
__global__ __launch_bounds__(64) void router_finalize(
    const float* __restrict__ ws_psum,
    const float* __restrict__ ws_cnt,
    int nwg, float num_tokens,
    float* __restrict__ out_aux)
{
    __shared__ float contrib[NEXP];
    const int e = threadIdx.x;

    float ps = 0.f, cs = 0.f;
    for (int w = 0; w < nwg; ++w) {
        ps += ws_psum[(size_t)w * NEXP + e];
        cs += ws_cnt [(size_t)w * NEXP + e];
    }
    const float f = cs / (num_tokens * (float)TOPK);  // fraction routed to e
    const float p = ps / num_tokens;                  // mean router prob of e
    contrib[e] = f * p;
    __syncthreads();

    if (e == 0) {
        float s = 0.f;
#pragma unroll
        for (int i = 0; i < NEXP; ++i) s += contrib[i];
        out_aux[0] = (float)NEXP * s;
    }
}

// ---------------------------------------------------------------------------
extern "C" void kernel_launch(void* const* d_in, const int* in_sizes, int n_in,
                              void* d_out, int out_size, void* d_ws, size_t ws_size,
                              hipStream_t stream) {
    const float* H = (const float*)d_in[0];   // (T, 4096) fp32
    const float* G = (const float*)d_in[1];   // (64, 4096) fp32

    const int tokens = in_sizes[0] / HID;     // 8192
    const int nwg    = tokens / TILE_T;       // 512

    float* out   = (float*)d_out;
    float* out_w = out;                                   // (T,8) weights
    float* out_i = out + (size_t)tokens * TOPK;           // (T,8) indices (as floats)
    float* out_a = out + (size_t)tokens * TOPK * 2;       // aux scalar

    float* ws_psum = (float*)d_ws;                        // nwg*64 floats
    float* ws_cnt  = ws_psum + (size_t)nwg * NEXP;        // nwg*64 floats

    router_stage1<<<nwg, 128, 0, stream>>>(H, G, out_w, out_i, ws_psum, ws_cnt);
    router_finalize<<<1, NEXP, 0, stream>>>(ws_psum, ws_cnt, nwg, (float)tokens, out_a);
}